// LDS_26276609917180
// MI455X (gfx1250) — compile-verified
//
#include <hip/hip_runtime.h>
#include <math.h>

typedef __attribute__((ext_vector_type(2))) float v2f;
typedef __attribute__((ext_vector_type(8))) float v8f;

#define NT 8      // ntrials
#define BD 64     // latent dim b
#define TD 512    // T
#define ND 512    // N neurons
#define XD 128    // x_dim
#define MC 32     // n_mc
#define JITTER 1e-6f

// ---------------- workspace layout (bytes) ----------------
// 0     : double acc[32]  (0..7: mc sum, 8..15: lgamma sum, 16..23: quad_trans, 24..31: quad0)
// 256   : float cholR[128]
// 768   : float logdets[2]   (0: logdet_q, 1: logdet_0)
// 1024  : float Linv_q[64*64]
// 17408 : float Linv_0[64*64]
// 33792 : float muW[8*128*512]   (2 MB)

// ============================================================
// prep: chol_R, Cholesky of Q=BB^T+jI and Sigma0, tri-inverse, logdets
// ============================================================
__global__ void prep_kernel(const float* __restrict__ B, const float* __restrict__ S0h,
                            const float* __restrict__ lsx,
                            float* __restrict__ cholR, float* __restrict__ Linv_q,
                            float* __restrict__ Linv_0, float* __restrict__ logdets) {
  __shared__ float M[BD * BD];
  const int tid = threadIdx.x;  // 128 threads
  if (tid < XD) cholR[tid] = sqrtf(expf(2.f * lsx[tid]) + JITTER);

  for (int phase = 0; phase < 2; ++phase) {
    const float* src = (phase == 0) ? B : S0h;
    float* Linv = (phase == 0) ? Linv_q : Linv_0;
    // M = src*src^T + jitter*I
    for (int idx = tid; idx < BD * BD; idx += 128) {
      int r = idx >> 6, c = idx & 63;
      float s = 0.f;
      for (int k = 0; k < BD; ++k) s += src[r * BD + k] * src[c * BD + k];
      if (r == c) s += JITTER;
      M[idx] = s;
    }
    __syncthreads();
    // in-place Cholesky (lower)
    for (int j = 0; j < BD; ++j) {
      if (tid == 0) M[j * BD + j] = sqrtf(M[j * BD + j]);
      __syncthreads();
      if (tid > j && tid < BD) M[tid * BD + j] /= M[j * BD + j];
      __syncthreads();
      if (tid > j && tid < BD) {
        float lij = M[tid * BD + j];
        for (int c = j + 1; c <= tid; ++c) M[tid * BD + c] -= lij * M[c * BD + j];
      }
      __syncthreads();
    }
    if (tid == 0) {
      float ld = 0.f;
      for (int j = 0; j < BD; ++j) ld += logf(M[j * BD + j]);
      logdets[phase] = 2.f * ld;
    }
    // Linv column tid via forward substitution
    if (tid < BD) {
      const int j = tid;
      for (int i = 0; i < j; ++i) Linv[i * BD + j] = 0.f;
      for (int i = j; i < BD; ++i) {
        float s = (i == j) ? 1.f : 0.f;
        for (int k = j; k < i; ++k) s -= M[i * BD + k] * Linv[k * BD + j];
        Linv[i * BD + j] = s / M[i * BD + i];
      }
    }
    __syncthreads();
  }
}

// ============================================================
// mu[n,x,t] = sum_b W[x,b] * z[n,b,t]
// ============================================================
__global__ void mu_kernel(const float* __restrict__ W, const float* __restrict__ z,
                          float* __restrict__ muW) {
  const int t = blockIdx.x * 256 + threadIdx.x;  // 0..511
  const int x = blockIdx.y;                      // 0..127
  const int n = blockIdx.z;                      // 0..7
  float s = 0.f;
  const float* zb = z + ((size_t)n * BD) * TD + t;
  const float* wb = W + (size_t)x * BD;
#pragma unroll 8
  for (int b = 0; b < BD; ++b) s += wb[b] * zb[(size_t)b * TD];
  muW[((size_t)n * XD + x) * TD + t] = s;
}

// ============================================================
// sum of lgamma(Y+1) per trial
// ============================================================
__global__ void lgamma_kernel(const int* __restrict__ Y, double* __restrict__ acc) {
  __shared__ float red[256];
  const int trial = blockIdx.y;
  const size_t i = (size_t)blockIdx.x * 256 + threadIdx.x;
  float v = lgammaf((float)Y[(size_t)trial * ND * TD + i] + 1.0f);
  red[threadIdx.x] = v;
  __syncthreads();
  for (int s = 128; s > 0; s >>= 1) {
    if (threadIdx.x < s) red[threadIdx.x] += red[threadIdx.x + s];
    __syncthreads();
  }
  if (threadIdx.x == 0) atomicAdd(&acc[8 + trial], (double)red[0]);
}

// ============================================================
// transition + prior quadratic forms
// ============================================================
__global__ __launch_bounds__(256) void trans_kernel(
    const float* __restrict__ z, const float* __restrict__ A, const float* __restrict__ mu0,
    const float* __restrict__ Linv_q, const float* __restrict__ Linv_0,
    double* __restrict__ acc) {
  __shared__ float zc[BD * 66];  // columns t = tbase-1 .. tbase+63 (u = 0..64), pitch 66
  __shared__ float Al[BD * BD];
  __shared__ float Lq[BD * BD];
  __shared__ float vl[BD * 65];  // pitch 65
  __shared__ float red[256];
  const int tid = threadIdx.x;
  const int chunk = blockIdx.x;  // 0..7
  const int trial = blockIdx.y;  // 0..7
  const int tbase = chunk * 64;

  for (int idx = tid; idx < BD * 65; idx += 256) {
    int j = idx / 65, u = idx % 65;
    int t = tbase - 1 + u;
    zc[j * 66 + u] = (t >= 0) ? z[((size_t)trial * BD + j) * TD + t] : 0.f;
  }
  for (int idx = tid; idx < BD * BD; idx += 256) { Al[idx] = A[idx]; Lq[idx] = Linv_q[idx]; }
  __syncthreads();

  const int tq = tid & 63, jg = tid >> 6;
  // v[j][t] = z[j][t] - sum_k A[j][k] z[k][t-1]
  for (int j = jg; j < BD; j += 4) {
    float s = 0.f;
#pragma unroll 8
    for (int k = 0; k < BD; ++k) s += Al[j * BD + k] * zc[k * 66 + tq];
    vl[j * 65 + tq] = zc[j * 66 + tq + 1] - s;
  }
  __syncthreads();
  float local = 0.f;
  const bool tvalid = (tbase + tq) >= 1;
  for (int i = jg; i < BD; i += 4) {
    float y = 0.f;
#pragma unroll 8
    for (int k = 0; k < BD; ++k) y += Lq[i * BD + k] * vl[k * 65 + tq];
    if (tvalid) local += y * y;
  }
  red[tid] = local;
  __syncthreads();
  for (int s = 128; s > 0; s >>= 1) {
    if (tid < s) red[tid] += red[tid + s];
    __syncthreads();
  }
  if (tid == 0) atomicAdd(&acc[16 + trial], (double)red[0]);

  if (chunk == 0) {  // prior term on z[:,0]
    __syncthreads();
    if (tid < BD) vl[tid] = zc[tid * 66 + 1] - mu0[tid];
    __syncthreads();
    float l0 = 0.f;
    if (tid < BD) {
      float y = 0.f;
      for (int k = 0; k < BD; ++k) y += Linv_0[tid * BD + k] * vl[k];
      l0 = y * y;
    }
    red[tid] = l0;
    __syncthreads();
    for (int s = 128; s > 0; s >>= 1) {
      if (tid < s) red[tid] += red[tid + s];
      __syncthreads();
    }
    if (tid == 0) atomicAdd(&acc[24 + trial], (double)red[0]);
  }
}

// ============================================================
// MAIN: fused GEMM (fp32 WMMA) + Poisson LL reduction
// block tile: 128 neurons x 64 timesteps, one (m,trial); 8 waves,
// wave w owns neuron rows [16w,16w+16) and all 64 t (4 wmma tiles).
// ============================================================
#define XPITCH 134  // LDS pitch for X^T (even for b64 align, mod 64 = 6)

__global__ __launch_bounds__(256) void poisson_main(
    const float* __restrict__ C, const float* __restrict__ eps,
    const float* __restrict__ muW, const int* __restrict__ Y,
    const float* __restrict__ dvec, const float* __restrict__ cholR,
    double* __restrict__ acc) {
  __shared__ float Xl[64 * XPITCH];  // Xl[t][k] = cholR[k]*eps[k][t] + mu[k][t]
  __shared__ float red[256];
  const int tid = threadIdx.x;
  const int lane = tid & 31;
  const int wave = tid >> 5;
  const int t0 = blockIdx.x * 64;   // 8 tiles
  const int n0 = blockIdx.y * 128;  // 4 tiles
  const int mz = blockIdx.z;        // m*NT + trial, 256
  const int trial = mz & 7;

  // stage X^T strip (fused sample construction), coalesced on t
  {
    const int t = tid & 63;
    const int kb = tid >> 6;  // uniform within a wave
    const float* epsb = eps + ((size_t)mz * XD) * TD + t0 + t;
    const float* mub = muW + ((size_t)trial * XD) * TD + t0 + t;
    for (int k = kb; k < XD; k += 4) {
      Xl[t * XPITCH + k] = cholR[k] * epsb[(size_t)k * TD] + mub[(size_t)k * TD];
    }
  }

  // preload this wave's 16x128 strip of C into registers (A fragments)
  const int lr = lane & 15;
  const int hs = (lane < 16) ? 0 : 2;
  v2f areg[32];
  {
    const float* crow = C + (size_t)(n0 + wave * 16 + lr) * XD + hs;
#pragma unroll
    for (int kc = 0; kc < 32; ++kc) areg[kc] = *(const v2f*)(crow + 4 * kc);
  }
  __syncthreads();

  v8f acc0 = {}, acc1 = {}, acc2 = {}, acc3 = {};
  const float* brow0 = &Xl[(0 * 16 + lr) * XPITCH + hs];
  const float* brow1 = &Xl[(1 * 16 + lr) * XPITCH + hs];
  const float* brow2 = &Xl[(2 * 16 + lr) * XPITCH + hs];
  const float* brow3 = &Xl[(3 * 16 + lr) * XPITCH + hs];
#pragma unroll
  for (int kc = 0; kc < 32; ++kc) {
    v2f b0 = *(const v2f*)(brow0 + 4 * kc);
    v2f b1 = *(const v2f*)(brow1 + 4 * kc);
    v2f b2 = *(const v2f*)(brow2 + 4 * kc);
    v2f b3 = *(const v2f*)(brow3 + 4 * kc);
    acc0 = __builtin_amdgcn_wmma_f32_16x16x4_f32(false, areg[kc], false, b0, (short)0, acc0, false, false);
    acc1 = __builtin_amdgcn_wmma_f32_16x16x4_f32(false, areg[kc], false, b1, (short)0, acc1, false, false);
    acc2 = __builtin_amdgcn_wmma_f32_16x16x4_f32(false, areg[kc], false, b2, (short)0, acc2, false, false);
    acc3 = __builtin_amdgcn_wmma_f32_16x16x4_f32(false, areg[kc], false, b3, (short)0, acc3, false, false);
  }

  // epilogue: logit = D + d[n]; sum Y*logit - exp(logit)
  float lsum = 0.f;
  const int nb = n0 + wave * 16 + ((lane < 16) ? 0 : 8);
  const int tb = t0 + lr;
  const int* Yb = Y + ((size_t)trial * ND) * TD;
#pragma unroll
  for (int r = 0; r < 8; ++r) {
    const int neuron = nb + r;
    const float dv = dvec[neuron];
    const int* Yr = Yb + (size_t)neuron * TD + tb;
    float lg;
    lg = acc0[r] + dv; lsum += (float)Yr[0]  * lg - expf(lg);
    lg = acc1[r] + dv; lsum += (float)Yr[16] * lg - expf(lg);
    lg = acc2[r] + dv; lsum += (float)Yr[32] * lg - expf(lg);
    lg = acc3[r] + dv; lsum += (float)Yr[48] * lg - expf(lg);
  }
  red[tid] = lsum;
  __syncthreads();
  for (int s = 128; s > 0; s >>= 1) {
    if (tid < s) red[tid] += red[tid + s];
    __syncthreads();
  }
  if (tid == 0) atomicAdd(&acc[trial], (double)red[0]);
}

// ============================================================
// finalize: out[n] = mc/MC - lgam - 0.5(q0 + c + ld0) - 0.5(qt + (T-1)(c + ldq))
// ============================================================
__global__ void finalize_kernel(const double* __restrict__ acc,
                                const float* __restrict__ logdets,
                                float* __restrict__ out) {
  const int n = threadIdx.x;
  if (n < NT) {
    const double blog2pi = (double)BD * 1.8378770664093453;  // b*log(2*pi)
    double first = acc[n] / (double)MC - acc[8 + n];
    double ss = -0.5 * (acc[24 + n] + blog2pi + (double)logdets[1]);
    double sb = -0.5 * (acc[16 + n] + (double)(TD - 1) * (blog2pi + (double)logdets[0]));
    out[n] = (float)(first + ss + sb);
  }
}

extern "C" void kernel_launch(void* const* d_in, const int* in_sizes, int n_in,
                              void* d_out, int out_size, void* d_ws, size_t ws_size,
                              hipStream_t stream) {
  (void)in_sizes; (void)n_in; (void)out_size; (void)ws_size;
  const float* z    = (const float*)d_in[0];
  const int*   Y    = (const int*)d_in[1];
  const float* eps  = (const float*)d_in[2];
  const float* A    = (const float*)d_in[3];
  const float* C    = (const float*)d_in[4];
  const float* W    = (const float*)d_in[5];
  const float* B    = (const float*)d_in[6];
  const float* mu0  = (const float*)d_in[7];
  const float* S0h  = (const float*)d_in[8];
  const float* lsx  = (const float*)d_in[9];
  const float* dvec = (const float*)d_in[10];
  float* out = (float*)d_out;

  char* ws = (char*)d_ws;
  double* accd    = (double*)ws;                    // 32 doubles
  float*  cholR   = (float*)(ws + 256);
  float*  logdets = (float*)(ws + 768);
  float*  Linv_q  = (float*)(ws + 1024);
  float*  Linv_0  = (float*)(ws + 1024 + 16384);
  float*  muW     = (float*)(ws + 1024 + 32768);    // 8*128*512 floats

  hipMemsetAsync(accd, 0, 32 * sizeof(double), stream);
  prep_kernel<<<1, 128, 0, stream>>>(B, S0h, lsx, cholR, Linv_q, Linv_0, logdets);
  mu_kernel<<<dim3(2, XD, NT), 256, 0, stream>>>(W, z, muW);
  lgamma_kernel<<<dim3(ND * TD / 256, NT), 256, 0, stream>>>(Y, accd);
  trans_kernel<<<dim3(TD / 64, NT), 256, 0, stream>>>(z, A, mu0, Linv_q, Linv_0, accd);
  poisson_main<<<dim3(TD / 64, ND / 128, MC * NT), 256, 0, stream>>>(C, eps, muW, Y, dvec, cholR, accd);
  finalize_kernel<<<1, 32, 0, stream>>>(accd, logdets, out);
}